// DynamicLyotFilter_21457656611306
// MI455X (gfx1250) — compile-verified
//
#include <hip/hip_runtime.h>
#include <hip/hip_bf16.h>

typedef __attribute__((ext_vector_type(2))) float v2f;
typedef __attribute__((ext_vector_type(8))) float v8f;

constexpr int B_   = 16;
constexpr int H_   = 256;
constexpr int W_   = 256;
constexpr int CIN  = 3;
constexpr int CMID = 64;
constexpr int COUT = 3;
constexpr int A_   = 31;              // HSI bands
constexpr int AP   = 32;              // padded K for einsum
constexpr int HW   = H_ * W_;         // 65536
constexpr int TILES_PER_B = HW / 16;  // 4096; 8 waves/block -> 512 blocks/batch
constexpr int KC   = 27;              // conv reduction dim 3*3*3
constexpr int KCP  = 28;              // padded to multiple of 4
constexpr int PCOL = 130;             // patch cols: 128 pixels + 2 halo
constexpr int PSZ  = 3 * PCOL * CIN;  // 1170 floats
constexpr int SRFS = 34;              // LDS stride for einsum B (banks + 8B align)
constexpr float BN_EPS = 1e-5f;

// ---------------------------------------------------------------------------
// Stage conv weights into LDS, K-padded 27->28 with zeros.
// ---------------------------------------------------------------------------
__device__ __forceinline__ void load_weights_lds(const float* __restrict__ conv_w,
                                                 float* wLds) {
  for (int i = threadIdx.x; i < CMID * KCP; i += blockDim.x) {
    const int o = i / KCP;
    const int k = i - o * KCP;
    wLds[i] = (k < KC) ? conv_w[o * KC + k] : 0.0f;
  }
}

// ---------------------------------------------------------------------------
// Stage the block's 3x130x3 input patch (SAME zero padding applied here).
// Block = 128 consecutive pixels = half of one image row.
// ---------------------------------------------------------------------------
__device__ __forceinline__ void load_patch_lds(const float* __restrict__ x,
                                               int b, int h, int c0,
                                               float* xs) {
  const long xb = (long)b * HW * CIN;
  for (int i = threadIdx.x; i < PSZ; i += blockDim.x) {
    const int dh  = i / (PCOL * CIN);
    const int r   = i - dh * (PCOL * CIN);
    const int col = r / CIN;
    const int ci  = r - col * CIN;
    const int hh = h + dh - 1;
    const int ww = c0 + col - 1;
    float v = 0.0f;
    if (hh >= 0 && hh < H_ && ww >= 0 && ww < W_)
      v = x[xb + ((long)hh * W_ + ww) * CIN + ci];
    xs[i] = v;
  }
}

// ---------------------------------------------------------------------------
// Implicit-GEMM 3x3 conv tile from the LDS patch: one wave does 16 pixels x
// 64 channels = 7 K-chunks x 4 channel groups = 28 v_wmma_f32_16x16x4_f32.
// A fetch per slot: both khalf candidates decode to compile-time LDS offsets;
// one cndmask selects — no masks, no branches, boundary already in the patch.
// k=27 pad slot is clamped to k=26; its product is killed by wLds' zero pad.
// ---------------------------------------------------------------------------
__device__ __forceinline__ void conv_tile(const float* __restrict__ xs,
                                          const float* __restrict__ wLds,
                                          int wid, int lane, v8f acc[4]) {
  const int m = lane & 15;
  const int khalf = lane >> 4;
  const int colb = (wid * 16 + m) * CIN;   // patch base for this pixel (kw=-1)

  const v8f z = {0.f, 0.f, 0.f, 0.f, 0.f, 0.f, 0.f, 0.f};
#pragma unroll
  for (int g = 0; g < 4; ++g) acc[g] = z;

#pragma unroll
  for (int c = 0; c < 7; ++c) {
    v2f a;
#pragma unroll
    for (int r = 0; r < 2; ++r) {
      const int k0  = 4 * c + r;                         // khalf=0, always <27
      const int ci0 = k0 / 9, q0 = k0 % 9, kh0 = q0 / 3, kw0 = q0 % 3;
      const int off0 = (kh0 * PCOL + kw0) * CIN + ci0;   // constant
      const int k1  = 4 * c + 2 + r;
      const int kv1 = (k1 < KC) ? k1 : KC - 1;           // clamp pad slot
      const int ci1 = kv1 / 9, q1 = kv1 % 9, kh1 = q1 / 3, kw1 = q1 % 3;
      const int off1 = (kh1 * PCOL + kw1) * CIN + ci1;   // constant
      a[r] = xs[colb + (khalf ? off1 : off0)];           // 1 cndmask + ds_load
    }
#pragma unroll
    for (int g = 0; g < 4; ++g) {
      const int n = g * 16 + m;                          // output channel
      v2f bm;
      bm[0] = wLds[n * KCP + 4 * c + 2 * khalf + 0];     // 8B-aligned pair
      bm[1] = wLds[n * KCP + 4 * c + 2 * khalf + 1];
      acc[g] = __builtin_amdgcn_wmma_f32_16x16x4_f32(
          false, a, false, bm, (short)0, acc[g], false, false);
    }
  }
}

// ---------------------------------------------------------------------------
// Pass 1: conv -> per-channel sum & sum-of-squares (batch statistics).
// conv bias omitted: it cancels exactly through BatchNorm's mean subtraction.
// ---------------------------------------------------------------------------
__global__ void k_conv_stats(const float* __restrict__ x,
                             const float* __restrict__ conv_w,
                             float* __restrict__ gsum,
                             float* __restrict__ gsumsq) {
  __shared__ float wLds[CMID * KCP];
  __shared__ float xs[PSZ];
  __shared__ float sL[CMID];
  __shared__ float qL[CMID];

  const int T0 = blockIdx.x * 8;
  const int b  = T0 / TILES_PER_B;
  const int pbase = (T0 - b * TILES_PER_B) * 16;   // 128-aligned pixel offset
  const int h  = pbase >> 8;
  const int c0 = pbase & 255;

  load_weights_lds(conv_w, wLds);
  load_patch_lds(x, b, h, c0, xs);
  if (threadIdx.x < CMID) { sL[threadIdx.x] = 0.0f; qL[threadIdx.x] = 0.0f; }
  __syncthreads();

  const int wid  = threadIdx.x >> 5;
  const int lane = threadIdx.x & 31;
  v8f acc[4];
  conv_tile(xs, wLds, wid, lane, acc);

  const int m = lane & 15;
#pragma unroll
  for (int g = 0; g < 4; ++g) {
    float s = 0.0f, q = 0.0f;
#pragma unroll
    for (int r = 0; r < 8; ++r) { const float v = acc[g][r]; s += v; q += v * v; }
    atomicAdd(&sL[g * 16 + m], s);   // ds_add_f32
    atomicAdd(&qL[g * 16 + m], q);
  }
  __syncthreads();
  if (threadIdx.x < CMID) {
    atomicAdd(&gsum[threadIdx.x], sL[threadIdx.x]);
    atomicAdd(&gsumsq[threadIdx.x], qL[threadIdx.x]);
  }
}

// Fold BN into y*scale + shift per channel.
__global__ void k_bnstats(const float* __restrict__ gsum,
                          const float* __restrict__ gsumsq,
                          const float* __restrict__ bn_w,
                          const float* __restrict__ bn_b,
                          float* __restrict__ scale,
                          float* __restrict__ shift) {
  const int c = threadIdx.x;
  if (c < CMID) {
    const float invN = 1.0f / (float)(B_ * HW);
    const float mu  = gsum[c] * invN;
    const float var = gsumsq[c] * invN - mu * mu;
    const float rs  = rsqrtf(var + BN_EPS);
    const float sc  = rs * bn_w[c];
    scale[c] = sc;
    shift[c] = bn_b[c] - mu * sc;
  }
}

// ---------------------------------------------------------------------------
// Pass 2: conv -> BN -> ReLU -> per-(batch,channel) spatial sum (avgpool).
// ---------------------------------------------------------------------------
__global__ void k_conv_feat(const float* __restrict__ x,
                            const float* __restrict__ conv_w,
                            const float* __restrict__ scale,
                            const float* __restrict__ shift,
                            float* __restrict__ featsum) {
  __shared__ float wLds[CMID * KCP];
  __shared__ float xs[PSZ];
  __shared__ float sL[CMID];

  const int T0 = blockIdx.x * 8;
  const int b  = T0 / TILES_PER_B;                 // whole block shares b
  const int pbase = (T0 - b * TILES_PER_B) * 16;
  const int h  = pbase >> 8;
  const int c0 = pbase & 255;

  load_weights_lds(conv_w, wLds);
  load_patch_lds(x, b, h, c0, xs);
  if (threadIdx.x < CMID) sL[threadIdx.x] = 0.0f;
  __syncthreads();

  const int wid  = threadIdx.x >> 5;
  const int lane = threadIdx.x & 31;
  v8f acc[4];
  conv_tile(xs, wLds, wid, lane, acc);

  const int m = lane & 15;
#pragma unroll
  for (int g = 0; g < 4; ++g) {
    const int ch = g * 16 + m;
    const float sc = scale[ch];
    const float sh = shift[ch];
    float s = 0.0f;
#pragma unroll
    for (int r = 0; r < 8; ++r) s += fmaxf(acc[g][r] * sc + sh, 0.0f);
    atomicAdd(&sL[ch], s);
  }
  __syncthreads();
  if (threadIdx.x < CMID)
    atomicAdd(&featsum[b * CMID + threadIdx.x], sL[threadIdx.x]);
}

// ---------------------------------------------------------------------------
// Head: avgpool -> FC(64->5) -> minmax norm -> sort, drop ends -> V -> SRF.
// Emits a zero-padded 16x32 B-matrix per batch (srfPad).
// ---------------------------------------------------------------------------
__global__ void k_head(const float* __restrict__ featsum,
                       const float* __restrict__ fc_w,
                       const float* __restrict__ fc_b,
                       float* __restrict__ srfPad) {
  const int b = threadIdx.x;
  if (b >= B_) return;
  float feat[CMID];
  const float inv = 1.0f / (float)HW;
  for (int c = 0; c < CMID; ++c) feat[c] = featsum[b * CMID + c] * inv;
  float raw[5];
  for (int j = 0; j < 5; ++j) {
    float s = fc_b[j];
    for (int c = 0; c < CMID; ++c) s += feat[c] * fc_w[j * CMID + c];
    raw[j] = s;
  }
  float mn = raw[0], mx = raw[0];
  for (int j = 1; j < 5; ++j) { mn = fminf(mn, raw[j]); mx = fmaxf(mx, raw[j]); }
  float nrm[5];
  for (int j = 0; j < 5; ++j) nrm[j] = (raw[j] - mn) / (mx - mn + 1e-8f);
  for (int i = 0; i < 5; ++i)
    for (int j = i + 1; j < 5; ++j)
      if (nrm[j] < nrm[i]) { const float t = nrm[i]; nrm[i] = nrm[j]; nrm[j] = t; }
  float num[COUT];
  for (int cc = 0; cc < COUT; ++cc) {
    const float V = nrm[cc + 1] * 10.0f;
    const float cosp = cosf(6.28318530717958647692f * (-0.01f) / V);
    num[cc] = 0.5f - 0.5f * cosp;
  }
  for (int n = 0; n < 16; ++n) {
    for (int k = 0; k < AP; ++k) {
      float v = 0.0f;
      if (n < COUT && k < A_) {
        const float band = 400.0f + 300.0f * (float)k / 31.0f;
        v = num[n] / (band * 1e-6f);
      }
      srfPad[(b * 16 + n) * AP + k] = v;
    }
  }
}

// ---------------------------------------------------------------------------
// Einsum 'bca,bpqa->bpqc': per wave, 16 pixels x 16 N-cols (3 live), K=31->32.
// B operand staged to LDS with stride 34 (conflict-free, 8B-aligned b64 pairs);
// hot loop per chunk = 1 ds_load + 2 global b32 + 1 wmma. Bandwidth-bound on
// x_hsi (130 MB -> ~6 us at 23.3 TB/s); A's k=31 garbage killed by B zero pad.
// ---------------------------------------------------------------------------
__global__ void k_einsum(const float* __restrict__ x_hsi,
                         const float* __restrict__ srfPad,
                         float* __restrict__ out) {
  __shared__ float xsrf[16 * SRFS];
  const int wid  = threadIdx.x >> 5;
  const int lane = threadIdx.x & 31;
  const int T  = blockIdx.x * 8 + wid;
  const int b  = T / TILES_PER_B;              // whole block shares b
  const int pb = (T - b * TILES_PER_B) * 16;

  for (int i = threadIdx.x; i < 16 * AP; i += blockDim.x) {
    const int n = i >> 5, k = i & 31;
    xsrf[n * SRFS + k] = srfPad[b * (16 * AP) + i];
  }
  __syncthreads();

  const int m     = lane & 15;   // A: pixel row; B: channel col
  const int khalf = lane >> 4;
  const float* hb = x_hsi + ((long)b * HW + pb + m) * A_;
  const float* sp = xsrf + m * SRFS + 2 * khalf;

  v8f acc = {0.f, 0.f, 0.f, 0.f, 0.f, 0.f, 0.f, 0.f};
#pragma unroll
  for (int c = 0; c < 8; ++c) {
    const int kb = 4 * c + 2 * khalf;
    v2f a, bm;
    a[0]  = hb[(kb + 0 < A_) ? kb + 0 : A_ - 1];   // clamp, zero-killed by B
    a[1]  = hb[(kb + 1 < A_) ? kb + 1 : A_ - 1];
    bm[0] = sp[4 * c + 0];                         // ds_load_b64 pair
    bm[1] = sp[4 * c + 1];
    acc = __builtin_amdgcn_wmma_f32_16x16x4_f32(
        false, a, false, bm, (short)0, acc, false, false);
  }
  if (m < COUT) {
#pragma unroll
    for (int r = 0; r < 8; ++r) {
      const int p = pb + r + 8 * khalf;            // D: M = r + 8*khalf
      out[((long)b * HW + p) * COUT + m] = acc[r];
    }
  }
}

__global__ void k_zero(float* __restrict__ ws, int n) {
  const int i = blockIdx.x * blockDim.x + threadIdx.x;
  if (i < n) ws[i] = 0.0f;
}

// ---------------------------------------------------------------------------
// Workspace layout (floats): gsum[64] | gsumsq[64] | scale[64] | shift[64] |
//                            featsum[16*64] | srfPad[16*16*32]
// ---------------------------------------------------------------------------
extern "C" void kernel_launch(void* const* d_in, const int* in_sizes, int n_in,
                              void* d_out, int out_size, void* d_ws, size_t ws_size,
                              hipStream_t stream) {
  (void)in_sizes; (void)n_in; (void)out_size; (void)ws_size;
  const float* x      = (const float*)d_in[0];
  const float* x_hsi  = (const float*)d_in[1];
  const float* conv_w = (const float*)d_in[2];
  // d_in[3] = conv_b: unused — cancels exactly through BatchNorm.
  const float* bn_w   = (const float*)d_in[4];
  const float* bn_b   = (const float*)d_in[5];
  const float* fc_w   = (const float*)d_in[6];
  const float* fc_b   = (const float*)d_in[7];
  float* out = (float*)d_out;

  float* ws      = (float*)d_ws;
  float* gsum    = ws;
  float* gsumsq  = ws + 64;
  float* scale   = ws + 128;
  float* shift   = ws + 192;
  float* featsum = ws + 256;          // 1024 floats
  float* srfPad  = ws + 1280;         // 16*16*32 = 8192 floats

  k_zero<<<5, 256, 0, stream>>>(ws, 1280);   // accumulators must start at 0

  const int blocks = (B_ * HW) / (16 * 8);   // 8192 blocks, 8 waves each
  k_conv_stats<<<blocks, 256, 0, stream>>>(x, conv_w, gsum, gsumsq);
  k_bnstats<<<1, 64, 0, stream>>>(gsum, gsumsq, bn_w, bn_b, scale, shift);
  k_conv_feat<<<blocks, 256, 0, stream>>>(x, conv_w, scale, shift, featsum);
  k_head<<<1, 32, 0, stream>>>(featsum, fc_w, fc_b, srfPad);
  k_einsum<<<blocks, 256, 0, stream>>>(x_hsi, srfPad, out);
}